// glim_28776280883320
// MI455X (gfx1250) — compile-verified
//
#include <hip/hip_runtime.h>

// GS=32, PARAM=64, B=16 -> n = 2048; per-batch A = kron(I,S) + kron(M_b,N).
// Pipeline: build A in ws -> in-place blocked Gauss-Jordan inverse (32 steps,
// 64x64 blocks, V_WMMA_F32_16X16X4_F32 for all block GEMMs) -> two clamped
// matvecs (val -> ws -> d_out). All launches stream-ordered (graph-safe).
// Pivot-row blocks are double-buffered into LDS with GLOBAL_LOAD_ASYNC_TO_LDS
// (ASYNCcnt) issued via inline asm; S_WAIT_ASYNCCNT via builtin/asm.

#define NN     2048   // matrix dim
#define BS     64     // block size
#define NB     32     // blocks per dim
#define NBATCH 16
#define LSTR   68     // LDS row stride (floats): 68%64==4 -> conflict-free tiles

typedef __attribute__((ext_vector_type(2))) float v2f;
typedef __attribute__((ext_vector_type(8))) float v8f;

#if defined(__gfx1250__)
#define HAVE_ASYNC 1
#else
#define HAVE_ASYNC 0
#endif

#if HAVE_ASYNC
template <int Np>
__device__ __forceinline__ void waitAsync() {
#if __has_builtin(__builtin_amdgcn_s_wait_asynccnt)
    __builtin_amdgcn_s_wait_asynccnt(Np);
#else
    asm volatile("s_wait_asynccnt %0" :: "i"(Np) : "memory");
#endif
}
#endif

// cooperative 64x64 block copy: global (stride NN) -> LDS (stride LSTR)
__device__ __forceinline__ void ldBlock(float* dst, const float* src, int tid, float scale) {
#pragma unroll
    for (int it = 0; it < 4; ++it) {
        int v = tid + it * 256;
        int r = v >> 4, c = (v & 15) * 4;
        float4 t = *(const float4*)(src + (size_t)r * NN + c);
        float4 o; o.x = t.x * scale; o.y = t.y * scale; o.z = t.z * scale; o.w = t.w * scale;
        *(float4*)(dst + r * LSTR + c) = o;
    }
}

#if HAVE_ASYNC
// async 64x64 block fill: 4 x b128 per lane -> ASYNCcnt += 4 per wave.
// VDST = LDS byte offset (flat addr[31:0], ISA 10.2); VADDR = 64-bit global.
__device__ __forceinline__ void asyncLdBlock(float* dst, const float* src, int tid) {
#pragma unroll
    for (int it = 0; it < 4; ++it) {
        int v = tid + it * 256;
        int r = v >> 4, c = (v & 15) * 4;
        const float* g = src + (size_t)r * NN + c;
        unsigned l = (unsigned)(unsigned long long)(dst + r * LSTR + c);
        asm volatile("global_load_async_to_lds_b128 %0, %1, off"
                     :: "v"(l), "v"(g)
                     : "memory");
    }
}
#endif

__device__ __forceinline__ void stBlock(float* dst, const float* srcLds, int tid) {
#pragma unroll
    for (int it = 0; it < 4; ++it) {
        int v = tid + it * 256;
        int r = v >> 4, c = (v & 15) * 4;
        *(float4*)(dst + (size_t)r * NN + c) = *(const float4*)(srcLds + r * LSTR + c);
    }
}

// acc += Asm(64x64) @ Bsm(64x64), 16x16 tile (tm,tn); 16 steps of K=4 WMMA.
// A-frag 16x4: lanes 0-15 hold M, VGPR0={K0,K2}, VGPR1={K1,K3}; B 4x16 mirrored.
__device__ __forceinline__ v8f wmmaTile(const float* Asm, const float* Bsm,
                                        v8f acc, int tm, int tn, int lane) {
    const int m  = lane & 15;
    const int hi = lane >> 4;
#pragma unroll
    for (int s = 0; s < 16; ++s) {
        const int k0 = s * 4 + 2 * hi;
        v2f a, b;
        a[0] = Asm[(tm * 16 + m) * LSTR + k0];
        a[1] = Asm[(tm * 16 + m) * LSTR + k0 + 1];
        b[0] = Bsm[(k0)     * LSTR + tn * 16 + m];
        b[1] = Bsm[(k0 + 1) * LSTR + tn * 16 + m];
        acc = __builtin_amdgcn_wmma_f32_16x16x4_f32(false, a, false, b,
                                                    (short)0, acc, false, false);
    }
    return acc;
}

// A[b, i*64+k, j*64+l] = mat[b,i,j]*N[k,l] + (i==j)*S[k,l]; one float4/thread
__global__ __launch_bounds__(256) void build_kernel(const float* __restrict__ mat,
                                                    const float* __restrict__ S,
                                                    const float* __restrict__ Ng,
                                                    float* __restrict__ A) {
    const size_t idx = (size_t)blockIdx.x * 256 + threadIdx.x;
    const int c4 = (int)(idx & 511);
    const size_t rem = idx >> 9;
    const int r = (int)(rem & 2047);
    const int b = (int)(rem >> 11);
    const int i = r >> 6, kk = r & 63;
    const int c = c4 * 4, j = c >> 6, l = c & 63;
    const float mv = mat[(b * 32 + i) * 32 + j];
    float4 nv = *(const float4*)(Ng + kk * 64 + l);
    float4 o; o.x = mv * nv.x; o.y = mv * nv.y; o.z = mv * nv.z; o.w = mv * nv.w;
    if (i == j) {
        float4 sv = *(const float4*)(S + kk * 64 + l);
        o.x += sv.x; o.y += sv.y; o.z += sv.z; o.w += sv.w;
    }
    *(float4*)(A + (size_t)b * NN * NN + (size_t)r * NN + c) = o;
}

// Step k part 1 (one WG per batch): Dinv = inv(A[k][k]) in LDS, then
// A[k][j] <- Dinv @ A[k][j] for j != k, and A[k][k] <- Dinv.
__global__ __launch_bounds__(256) void pivot_kernel(float* __restrict__ A, int k) {
    const int b = blockIdx.x;
    float* Ab = A + (size_t)b * NN * NN;
    float* rowK = Ab + (size_t)(k * BS) * NN;
    __shared__ float Ds[BS * LSTR];
#if HAVE_ASYNC
    __shared__ float Gs2[2][BS * LSTR];
#else
    __shared__ float Gs[BS * LSTR];
#endif
    const int tid = threadIdx.x;
    const int lane = tid & 31, w = tid >> 5;
    ldBlock(Ds, rowK + k * BS, tid, 1.0f);
#if HAVE_ASYNC
    asyncLdBlock(Gs2[0], rowK, tid);     // overlap j=0 fetch with pivot inversion
#endif
    __syncthreads();
    // in-place scalar Gauss-Jordan inverse of the 64x64 pivot (no pivoting)
    const int ri = tid >> 2, c0 = (tid & 3) * 16;
    for (int t = 0; t < BS; ++t) {
        const float p = 1.0f / Ds[t * LSTR + t];
        const float f = Ds[ri * LSTR + t];
        __syncthreads();
        if (ri == t) {
#pragma unroll
            for (int c = 0; c < 16; ++c) {
                const int cc = c0 + c;
                const float v = (cc == t) ? 1.0f : Ds[t * LSTR + cc];
                Ds[t * LSTR + cc] = v * p;
            }
        }
        __syncthreads();
        if (ri != t) {
#pragma unroll
            for (int c = 0; c < 16; ++c) {
                const int cc = c0 + c;
                const float rv  = Ds[t * LSTR + cc];
                const float cur = (cc == t) ? 0.0f : Ds[ri * LSTR + cc];
                Ds[ri * LSTR + cc] = cur - f * rv;
            }
        }
        __syncthreads();
    }
    // scale pivot block-row
    for (int j = 0; j < NB; ++j) {
        float* Gg = rowK + j * BS;
        __syncthreads();                       // prior-j consumers done
#if HAVE_ASYNC
        if (j + 1 < NB) {
            asyncLdBlock(Gs2[(j + 1) & 1], rowK + (j + 1) * BS, tid);
            waitAsync<4>();                    // buffer j complete (in-order)
        } else {
            waitAsync<0>();
        }
        __syncthreads();
        const float* Gsj = Gs2[j & 1];
        if (j == k) { stBlock(Gg, Ds, tid); continue; }
#else
        if (j == k) { stBlock(Gg, Ds, tid); continue; }
        ldBlock(Gs, Gg, tid, 1.0f);
        __syncthreads();
        const float* Gsj = Gs;
#endif
#pragma unroll
        for (int tt = 0; tt < 2; ++tt) {
            const int t = w * 2 + tt, tm = t >> 2, tn = t & 3;
            const int nn = lane & 15, hi = lane >> 4;
            v8f acc = {};
            acc = wmmaTile(Ds, Gsj, acc, tm, tn, lane);
#pragma unroll
            for (int r = 0; r < 8; ++r)
                Gg[(size_t)(tm * 16 + r + 8 * hi) * NN + tn * 16 + nn] = acc[r];
        }
    }
}

// Step k part 2 (one WG per (batch, row i != k)):
//   F = A[i][k] (old); A[i][j] <- A[i][j] - F @ A[k][j]  (j==k accum from 0
//   gives A[i][k] = -F @ Dinv, completing in-place Gauss-Jordan).
__global__ __launch_bounds__(256) void elim_kernel(float* __restrict__ A, int k) {
    const int b = blockIdx.y;
    int i = blockIdx.x; if (i >= k) ++i;   // gridDim.x = NB-1, skip pivot row
    float* Ab = A + (size_t)b * NN * NN;
    const float* rowK = Ab + (size_t)(k * BS) * NN;
    __shared__ float Fs[BS * LSTR];
#if HAVE_ASYNC
    __shared__ float Rs2[2][BS * LSTR];
#else
    __shared__ float Rs[BS * LSTR];
#endif
    const int tid = threadIdx.x;
    const int lane = tid & 31, w = tid >> 5;
#if HAVE_ASYNC
    asyncLdBlock(Rs2[0], rowK, tid);       // overlap j=0 fetch with F staging
#endif
    ldBlock(Fs, Ab + (size_t)(i * BS) * NN + k * BS, tid, -1.0f);  // stage -F
    for (int j = 0; j < NB; ++j) {
        __syncthreads();                   // prior-j consumers done, buffers free
#if HAVE_ASYNC
        if (j + 1 < NB) {
            asyncLdBlock(Rs2[(j + 1) & 1], rowK + (j + 1) * BS, tid);
            waitAsync<4>();                // buffer j complete (in-order)
        } else {
            waitAsync<0>();
        }
        __syncthreads();
        const float* Rsj = Rs2[j & 1];
#else
        const float* Rg = rowK + j * BS;
        ldBlock(Rs, Rg, tid, 1.0f);
        if (j + 1 < NB && tid < BS)        // pull next pivot-row block into cache
            __builtin_prefetch(Rg + BS + (size_t)tid * NN, 0, 1);
        __syncthreads();
        const float* Rsj = Rs;
#endif
        float* Cg = Ab + (size_t)(i * BS) * NN + j * BS;
#pragma unroll
        for (int tt = 0; tt < 2; ++tt) {
            const int t = w * 2 + tt, tm = t >> 2, tn = t & 3;
            const int nn = lane & 15, hi = lane >> 4;
            v8f acc = {};
            if (j != k) {
#pragma unroll
                for (int r = 0; r < 8; ++r)
                    acc[r] = Cg[(size_t)(tm * 16 + r + 8 * hi) * NN + tn * 16 + nn];
            }
            acc = wmmaTile(Fs, Rsj, acc, tm, tn, lane);   // acc += (-F) @ R
#pragma unroll
            for (int r = 0; r < 8; ++r)
                Cg[(size_t)(tm * 16 + r + 8 * hi) * NN + tn * 16 + nn] = acc[r];
        }
    }
}

// vout = clamp(P @ vin, -1, 1); one wave per output row, wave32 xor-reduction
__global__ __launch_bounds__(256) void matvec_kernel(const float* __restrict__ P,
                                                     const float* __restrict__ vin,
                                                     float* __restrict__ vout) {
    const int b = blockIdx.y;
    __shared__ float vs[NN];
    const float* Pb = P + (size_t)b * NN * NN;
    for (int idx = threadIdx.x; idx < NN; idx += 256) vs[idx] = vin[b * NN + idx];
    __syncthreads();
    const int w = threadIdx.x >> 5, lane = threadIdx.x & 31;
    const int row = blockIdx.x * 8 + w;
    const float* Pr = Pb + (size_t)row * NN;
    float s = 0.0f;
    for (int t = lane; t < NN; t += 32) s += Pr[t] * vs[t];
#pragma unroll
    for (int off = 16; off > 0; off >>= 1) s += __shfl_xor(s, off, 32);
    if (lane == 0) vout[b * NN + row] = fminf(1.0f, fmaxf(-1.0f, s));
}

extern "C" void kernel_launch(void* const* d_in, const int* in_sizes, int n_in,
                              void* d_out, int out_size, void* d_ws, size_t ws_size,
                              hipStream_t stream) {
    (void)in_sizes; (void)n_in; (void)out_size; (void)ws_size;
    const float* mat = (const float*)d_in[0];   // [16,32,32]
    const float* val = (const float*)d_in[1];   // [16,32,64]
    const float* S   = (const float*)d_in[2];   // [64,64]
    const float* Ng  = (const float*)d_in[3];   // [64,64]
    float* A    = (float*)d_ws;                                             // 256 MiB
    float* vtmp = (float*)((char*)d_ws + (size_t)NBATCH * NN * NN * sizeof(float));

    build_kernel<<<(NBATCH * NN * (NN / 4)) / 256, 256, 0, stream>>>(mat, S, Ng, A);
    for (int k = 0; k < NB; ++k) {
        pivot_kernel<<<NBATCH, 256, 0, stream>>>(A, k);
        elim_kernel<<<dim3(NB - 1, NBATCH), 256, 0, stream>>>(A, k);
    }
    matvec_kernel<<<dim3(NN / 8, NBATCH), 256, 0, stream>>>(A, val, vtmp);
    matvec_kernel<<<dim3(NN / 8, NBATCH), 256, 0, stream>>>(A, vtmp, (float*)d_out);
}